// SAGE_22170621182211
// MI455X (gfx1250) — compile-verified
//
#include <hip/hip_runtime.h>

// ---------------------------------------------------------------------------
// GraphSAGE 2-layer forward for MI455X (gfx1250, wave32, WMMA).
//   layer: out = relu( (segment_mean(h[src]->dst)) @ Wl + bl + h @ Wr )
// Fused as one K=256 GEMM per layer: [agg | h] @ [Wl ; Wr], fp32 WMMA
// (V_WMMA_F32_16X16X4_F32). W staged in LDS pre-paired along K so each
// B fragment is a single ds_load_b64 into an even-aligned VGPR pair.
// ---------------------------------------------------------------------------

typedef __attribute__((ext_vector_type(2))) float v2f;
typedef __attribute__((ext_vector_type(8))) float v8f;

#define D_FEAT 128

// ---------------------------- utility kernels ------------------------------

__global__ void zero_f32(float* __restrict__ p, int n) {
    int i = blockIdx.x * blockDim.x + threadIdx.x;
    if (i < n) p[i] = 0.0f;
}

__global__ void edge_count(const int* __restrict__ dst, float* __restrict__ cnt, int E) {
    int e = blockIdx.x * blockDim.x + threadIdx.x;
    if (e < E) {
        __hip_atomic_fetch_add(&cnt[dst[e]], 1.0f,
                               __ATOMIC_RELAXED, __HIP_MEMORY_SCOPE_AGENT);
    }
}

__global__ void make_inv(float* __restrict__ cnt, int N) {
    int i = blockIdx.x * blockDim.x + threadIdx.x;
    if (i < N) {
        float c = cnt[i];
        cnt[i] = 1.0f / fmaxf(c, 1.0f);   // becomes inv-degree
    }
}

// One wave per edge: lane l moves 4 consecutive floats (coalesced 512B row),
// scattered with native fp32 global atomics (no return -> STOREcnt path).
__global__ __launch_bounds__(256)
void scatter_add(const float* __restrict__ h, const int* __restrict__ src,
                 const int* __restrict__ dst, float* __restrict__ msg, int E) {
    int e = blockIdx.x * (blockDim.x >> 5) + (threadIdx.x >> 5);
    if (e >= E) return;
    int lane = threadIdx.x & 31;
    int s = src[e];
    int d = dst[e];
    const float4 v = *(const float4*)(h + (size_t)s * D_FEAT + lane * 4);
    float* p = msg + (size_t)d * D_FEAT + lane * 4;
    __hip_atomic_fetch_add(p + 0, v.x, __ATOMIC_RELAXED, __HIP_MEMORY_SCOPE_AGENT);
    __hip_atomic_fetch_add(p + 1, v.y, __ATOMIC_RELAXED, __HIP_MEMORY_SCOPE_AGENT);
    __hip_atomic_fetch_add(p + 2, v.z, __ATOMIC_RELAXED, __HIP_MEMORY_SCOPE_AGENT);
    __hip_atomic_fetch_add(p + 3, v.w, __ATOMIC_RELAXED, __HIP_MEMORY_SCOPE_AGENT);
}

// ------------------------------ fused GEMM ---------------------------------
// out[rowblock(128) x 128] = relu( (msg*inv) @ Wl + h @ Wr + bias )
// Per wave: A'[16x256] @ W'[256x128], K-chunked by 64. LDS holds the 64x128
// W-chunk as K-pairs: sW[(kk*128+col)*2 + {0,1}] = W[2kk{,+1}][col], so the
// 4x16 B fragment (VGPR0=K(2kh), VGPR1=K(2kh+1)) is one ds_load_b64.
__global__ __launch_bounds__(256)
void sage_gemm(const float* __restrict__ msg,  // N x 128 (neighbor sums)
               const float* __restrict__ inv,  // N      (1/degree)
               const float* __restrict__ h,    // N x 128 (self features)
               const float* __restrict__ Wl,   // 128 x 128 row-major
               const float* __restrict__ Wr,   // 128 x 128 row-major
               const float* __restrict__ bias, // 128
               float* __restrict__ out,        // N x 128
               int N) {
    __shared__ float sW[64 * 128];             // K-paired W chunk (32 KB)

    const int tid  = threadIdx.x;
    const int wave = tid >> 5;                 // 0..7 -> 16-row slab
    const int lane = tid & 31;
    const int m    = lane & 15;                // A row within tile / B col
    const int kh   = lane >> 4;                // K-half select

    const int rowbase = blockIdx.x * 128 + wave * 16;
    int arow = rowbase + m;
    if (arow >= N) arow = N - 1;               // clamp: keeps EXEC all-1s for WMMA
    const float rowscale = inv[arow];

    v8f acc[8];
    const v8f zero = {0.f, 0.f, 0.f, 0.f, 0.f, 0.f, 0.f, 0.f};
#pragma unroll
    for (int ct = 0; ct < 8; ++ct) acc[ct] = zero;

    // 4 K-chunks of 64: chunks 0,1 -> (msg, Wl); chunks 2,3 -> (h, Wr)
    for (int kc = 0; kc < 4; ++kc) {
        const float* Wsrc = (kc < 2) ? Wl : Wr;
        const float* Asrc = (kc < 2) ? msg : h;
        const float  s    = (kc < 2) ? rowscale : 1.0f;
        const int    k0g  = (kc & 1) * 64;     // chunk offset within 128

        __syncthreads();                       // previous chunk fully consumed
        // Cooperative K-pair interleave load: 1024 units, 4 per thread.
        // Unit u: kk = u>>5 (row pair), cq = u&31 (col quad).
#pragma unroll
        for (int i = 0; i < 4; ++i) {
            const int u  = tid + i * 256;
            const int kk = u >> 5;
            const int cq = (u & 31) * 4;
            const float4 r0 = *(const float4*)(Wsrc + (k0g + 2 * kk + 0) * 128 + cq);
            const float4 r1 = *(const float4*)(Wsrc + (k0g + 2 * kk + 1) * 128 + cq);
            float2* d2 = (float2*)&sW[(kk * 128 + cq) * 2];
            d2[0] = float2{r0.x, r1.x};
            d2[1] = float2{r0.y, r1.y};
            d2[2] = float2{r0.z, r1.z};
            d2[3] = float2{r0.w, r1.w};
        }
        __syncthreads();

        const float* arowp = Asrc + (size_t)arow * D_FEAT + k0g;
        for (int k = 0; k < 64; k += 4) {
            // A fragment 16x4: lane holds {A[m][k+2kh], A[m][k+2kh+1]}
            const float2 av = *(const float2*)(arowp + k + 2 * kh);
            v2f a;
            a.x = av.x * s;
            a.y = av.y * s;
            // B pair row for this lane's K-half
            const v2f* brow = (const v2f*)&sW[((k >> 1) + kh) * 256];
#pragma unroll
            for (int ct = 0; ct < 8; ++ct) {
                const v2f b = brow[ct * 16 + m];   // one ds_load_b64
                acc[ct] = __builtin_amdgcn_wmma_f32_16x16x4_f32(
                    /*neg_a=*/false, a, /*neg_b=*/false, b,
                    /*c_mod=*/(short)0, acc[ct],
                    /*reuse_a=*/false, /*reuse_b=*/false);
            }
        }
    }

    // epilogue: bias + relu + store (C layout: VGPR i -> M = i + 8*kh, N = m)
#pragma unroll
    for (int ct = 0; ct < 8; ++ct) {
        const int col = ct * 16 + m;
        const float bv = bias[col];
#pragma unroll
        for (int i = 0; i < 8; ++i) {
            const int row = rowbase + kh * 8 + i;
            if (row < N) {
                float v = acc[ct][i] + bv;
                out[(size_t)row * D_FEAT + col] = v > 0.f ? v : 0.f;
            }
        }
    }
}

// ------------------------------- launcher ----------------------------------

extern "C" void kernel_launch(void* const* d_in, const int* in_sizes, int n_in,
                              void* d_out, int out_size, void* d_ws, size_t ws_size,
                              hipStream_t stream) {
    const float* x   = (const float*)d_in[0];
    const int*   ei  = (const int*)d_in[1];   // [2, E] int32 (JAX x64 off)
    const float* Wl1 = (const float*)d_in[2];
    const float* bl1 = (const float*)d_in[3];
    const float* Wr1 = (const float*)d_in[4];
    const float* Wl2 = (const float*)d_in[5];
    const float* bl2 = (const float*)d_in[6];
    const float* Wr2 = (const float*)d_in[7];

    const int N = in_sizes[0] / D_FEAT;
    const int E = in_sizes[1] / 2;
    const int* src = ei;
    const int* dst = ei + E;

    // workspace: msg[N*128] | inv[N] | h1[N*128]
    float* msg = (float*)d_ws;
    float* inv = msg + (size_t)N * D_FEAT;
    float* h1  = inv + N;
    float* out = (float*)d_out;

    const int NF = N * D_FEAT;

    // in-degree -> inverse degree (shared by both layers)
    zero_f32  <<<(N + 255) / 256, 256, 0, stream>>>(inv, N);
    edge_count<<<(E + 255) / 256, 256, 0, stream>>>(dst, inv, E);
    make_inv  <<<(N + 255) / 256, 256, 0, stream>>>(inv, N);

    // ---- layer 1 ----
    zero_f32   <<<(NF + 255) / 256, 256, 0, stream>>>(msg, NF);
    scatter_add<<<(E + 7) / 8, 256, 0, stream>>>(x, src, dst, msg, E);
    sage_gemm  <<<(N + 127) / 128, 256, 0, stream>>>(msg, inv, x, Wl1, Wr1, bl1, h1, N);

    // ---- layer 2 ----
    zero_f32   <<<(NF + 255) / 256, 256, 0, stream>>>(msg, NF);
    scatter_add<<<(E + 7) / 8, 256, 0, stream>>>(h1, src, dst, msg, E);
    sage_gemm  <<<(N + 127) / 128, 256, 0, stream>>>(msg, inv, h1, Wl2, Wr2, bl2, out, N);
}